// GraphEncoder_10359461118641
// MI455X (gfx1250) — compile-verified
//
#include <hip/hip_runtime.h>
#include <hip/hip_bf16.h>
#include <stdint.h>

#define N_NODES   100000
#define N_EDGES   1600000
#define N_GRAPHS  64
#define EMBED     64
#define HIDDEN    128

typedef float v2f __attribute__((ext_vector_type(2)));
typedef float v8f __attribute__((ext_vector_type(8)));
typedef unsigned int u32x4 __attribute__((ext_vector_type(4)));
typedef unsigned int u32x8 __attribute__((ext_vector_type(8)));

// ---------------------------------------------------------------- utilities
__global__ void gcn_zerof(float* __restrict__ p, int n) {
    int i = blockIdx.x * blockDim.x + threadIdx.x;
    if (i < n) p[i] = 0.0f;
}

// deg[dst] += 1 per edge
__global__ void gcn_deg(const int* __restrict__ dst, float* __restrict__ deg) {
    int e = blockIdx.x * blockDim.x + threadIdx.x;
    if (e < N_EDGES) atomicAdd(&deg[dst[e]], 1.0f);
}

// dinv = rsqrt(deg + 1)  (in place)
__global__ void gcn_dinv(float* __restrict__ d) {
    int i = blockIdx.x * blockDim.x + threadIdx.x;
    if (i < N_NODES) d[i] = rsqrtf(d[i] + 1.0f);
}

// h[i][c] = emb[x[i]][c], c in [0,64); h has row stride 128
__global__ void gcn_embed(const int* __restrict__ x, const float* __restrict__ emb,
                          float* __restrict__ h) {
    int t = blockIdx.x * blockDim.x + threadIdx.x;   // N*64 threads
    int i = t >> 6, c = t & 63;
    h[i * HIDDEN + c] = emb[x[i] * EMBED + c];
}

// ------------------------------------------------------------- WMMA GEMM
// Y[rows,128] = A[rows,K] @ W[K,128]; A row stride 128; rows % 16 == 0.
// One 16x16 output tile per wave; 8 waves/block cover all 8 column tiles of
// one 16-row strip.
//  - W[K x 128] staged into LDS by the Tensor Data Mover (TENSORcnt).
//  - A 16-row strip (8 KB) staged into LDS with async-to-LDS DMA (ASYNCcnt),
//    removing the 8x redundant A reads across the block's waves.
__global__ __launch_bounds__(256)
void gcn_gemm_wmma(const float* __restrict__ A, const float* __restrict__ W,
                   float* __restrict__ Y, int K) {
    __shared__ float sW[HIDDEN * HIDDEN];   // 64 KB max, K*128 floats used
    __shared__ float sA[16 * HIDDEN];       // 8 KB: this block's 16-row strip
    const int tid = threadIdx.x;

    // ---- TDM: stage W[K x 128] into LDS (wave 0 issues one descriptor) ----
    if (tid < 32) {
        unsigned lds_base = (unsigned)(uintptr_t)(&sW[0]);
        unsigned long long ga = (unsigned long long)(uintptr_t)W;
        u32x4 g0;
        g0[0] = 0x1u;                                   // count=1, user D#
        g0[1] = lds_base;                               // lds_addr
        g0[2] = (unsigned)(ga & 0xFFFFFFFFu);           // global_addr[31:0]
        g0[3] = (unsigned)((ga >> 32) & 0x01FFFFFFu)    // global_addr[56:32]
              | (2u << 30);                             // type=2 (image)
        u32x8 g1;
        g1[0] = 0x00020000u;                 // data_size=2 (4 bytes), no mask
        g1[1] = ((unsigned)HIDDEN) << 16;    // tensor_dim0[15:0] = 128
        g1[2] = ((unsigned)K) << 16;         // dim0[31:16]=0 | tensor_dim1[15:0]=K
        g1[3] = ((unsigned)HIDDEN) << 16;    // dim1[31:16]=0 | tile_dim0=128
        g1[4] = (unsigned)K;                 // tile_dim1=K, tile_dim2=0
        g1[5] = (unsigned)HIDDEN;            // tensor_dim0_stride=128
        g1[6] = 0u;                          // stride hi / dim1_stride lo
        g1[7] = 0u;                          // dim1_stride hi
        asm volatile("tensor_load_to_lds %0, %1, null, null"
                     :: "s"(g0), "s"(g1) : "memory");
        __builtin_amdgcn_s_wait_tensorcnt(0);
    }

    // ---- async DMA: stage this block's 16x128 A strip into LDS ----
    {
        const float* astrip = A + (size_t)blockIdx.x * 16 * HIDDEN;
        for (int i = tid * 4; i < 16 * HIDDEN; i += 256 * 4) {
            unsigned lds_addr = (unsigned)(uintptr_t)(&sA[i]);
            unsigned long long gaddr = (unsigned long long)(uintptr_t)(astrip + i);
            asm volatile("global_load_async_to_lds_b128 %0, %1, off"
                         :: "v"(lds_addr), "v"(gaddr) : "memory");
        }
        asm volatile("s_wait_asynccnt 0" ::: "memory");
    }
    __syncthreads();

    const int wave = tid >> 5;        // column tile 0..7
    const int lane = tid & 31;
    const int half = lane >> 4;       // 0: K pair {0,1}; 1: K pair {2,3}
    const int lr   = lane & 15;
    const int col  = wave * 16 + lr;  // B/D column for this lane

    v8f acc = {};
    for (int k = 0; k < K; k += 4) {
        const int ka = k + half * 2;
        v2f a, b;
        a.x = sA[lr * HIDDEN + ka];
        a.y = sA[lr * HIDDEN + ka + 1];
        b.x = sW[ka * HIDDEN + col];
        b.y = sW[(ka + 1) * HIDDEN + col];
        acc = __builtin_amdgcn_wmma_f32_16x16x4_f32(
            /*neg_a=*/false, a, /*neg_b=*/false, b,
            /*c_mod=*/(short)0, acc, /*reuse_a=*/false, /*reuse_b=*/false);
    }

    // D layout: VGPR j -> M = half*8 + j, N = col
    const int orow = blockIdx.x * 16 + half * 8;
#pragma unroll
    for (int j = 0; j < 8; ++j)
        Y[(orow + j) * HIDDEN + col] = acc[j];
}

// ------------------------------------------------------------- GCN layer ops
// agg[i][c] = m[i][c] * dinv[i]^2   (self-loop term; also re-inits agg)
__global__ void gcn_selfinit(const float* __restrict__ m, const float* __restrict__ dinv,
                             float* __restrict__ agg) {
    int t = blockIdx.x * blockDim.x + threadIdx.x;   // N*128
    int i = t >> 7;
    float d = dinv[i];
    agg[t] = m[t] * d * d;
}

// agg[dst] += m[src] * dinv[src]*dinv[dst], 4 channels per thread
__global__ void gcn_scatter(const int* __restrict__ src, const int* __restrict__ dst,
                            const float* __restrict__ m, const float* __restrict__ dinv,
                            float* __restrict__ agg) {
    int t = blockIdx.x * blockDim.x + threadIdx.x;   // E*32
    int e  = t >> 5;
    int c4 = (t & 31) * 4;
    int s = src[e], d = dst[e];
    float coef = dinv[s] * dinv[d];
    const float4 v = *(const float4*)(m + s * HIDDEN + c4);
    float* base = agg + d * HIDDEN + c4;
    atomicAdd(base + 0, v.x * coef);
    atomicAdd(base + 1, v.y * coef);
    atomicAdd(base + 2, v.z * coef);
    atomicAdd(base + 3, v.w * coef);
}

// h = relu(agg + b)  (in place)
__global__ void gcn_bias_relu(float* __restrict__ agg, const float* __restrict__ b) {
    int t = blockIdx.x * blockDim.x + threadIdx.x;   // N*128
    float v = agg[t] + b[t & 127];
    agg[t] = v > 0.0f ? v : 0.0f;
}

// ------------------------------------------------------------- pooling
__global__ void gcn_pool(const int* __restrict__ batch, const float* __restrict__ h,
                         float* __restrict__ pooled) {
    int t = blockIdx.x * blockDim.x + threadIdx.x;   // N*128
    int i = t >> 7, c = t & 127;
    atomicAdd(&pooled[batch[i] * HIDDEN + c], h[t]);
}

__global__ void gcn_cnt(const int* __restrict__ batch, float* __restrict__ cnt) {
    int i = blockIdx.x * blockDim.x + threadIdx.x;
    if (i < N_NODES) atomicAdd(&cnt[batch[i]], 1.0f);
}

__global__ void gcn_avg(float* __restrict__ pooled, const float* __restrict__ cnt) {
    int t = blockIdx.x * blockDim.x + threadIdx.x;   // 64*128
    float c = cnt[t >> 7];
    pooled[t] = pooled[t] / (c < 1.0f ? 1.0f : c);
}

__global__ void gcn_outbias(const float* __restrict__ y, const float* __restrict__ bp,
                            float* __restrict__ out) {
    int t = blockIdx.x * blockDim.x + threadIdx.x;   // 64*128
    out[t] = y[t] + bp[t & 127];
}

// ---------------------------------------------------------------- launcher
extern "C" void kernel_launch(void* const* d_in, const int* in_sizes, int n_in,
                              void* d_out, int out_size, void* d_ws, size_t ws_size,
                              hipStream_t stream) {
    (void)in_sizes; (void)n_in; (void)out_size; (void)ws_size;
    const int*   x     = (const int*)d_in[0];
    const int*   ei    = (const int*)d_in[1];          // [2, E]
    const int*   batch = (const int*)d_in[2];
    const float* emb   = (const float*)d_in[3];
    const float* W1 = (const float*)d_in[4];  const float* b1 = (const float*)d_in[5];
    const float* W2 = (const float*)d_in[6];  const float* b2 = (const float*)d_in[7];
    const float* W3 = (const float*)d_in[8];  const float* b3 = (const float*)d_in[9];
    const float* Wp = (const float*)d_in[10]; const float* bp = (const float*)d_in[11];
    const int* src = ei;
    const int* dst = ei + N_EDGES;

    // workspace: dinv[N] | X[N*128] | Y[N*128] | pooled[64*128] | cnt[64]
    float* dinv   = (float*)d_ws;
    float* X      = dinv + N_NODES;
    float* Y      = X + (size_t)N_NODES * HIDDEN;
    float* pooled = Y + (size_t)N_NODES * HIDDEN;
    float* cnt    = pooled + N_GRAPHS * HIDDEN;

    const int NB_N    = (N_NODES + 255) / 256;          // 391
    const int NB_E    = N_EDGES / 256;                  // 6250
    const int NB_NE   = (N_NODES * EMBED) / 256;        // 25000
    const int NB_NH   = (N_NODES * HIDDEN) / 256;       // 50000
    const int NB_EDGE = (N_EDGES / 256) * 32;           // 200000
    const int NB_GEMM = N_NODES / 16;                   // 6250

    // degrees -> dinv
    gcn_zerof<<<NB_N, 256, 0, stream>>>(dinv, N_NODES);
    gcn_deg  <<<NB_E, 256, 0, stream>>>(dst, dinv);
    gcn_dinv <<<NB_N, 256, 0, stream>>>(dinv);

    // embedding gather into X (row stride 128, cols [0,64))
    gcn_embed<<<NB_NE, 256, 0, stream>>>(x, emb, X);

    // three GCN layers: Y = X@W; X = Y*dinv^2; X += scatter(Y); X = relu(X+b)
    const float* Ws[3] = {W1, W2, W3};
    const float* bs[3] = {b1, b2, b3};
    const int    Ks[3] = {EMBED, HIDDEN, HIDDEN};
    for (int l = 0; l < 3; ++l) {
        gcn_gemm_wmma<<<NB_GEMM, 256, 0, stream>>>(X, Ws[l], Y, Ks[l]);
        gcn_selfinit <<<NB_NH, 256, 0, stream>>>(Y, dinv, X);
        gcn_scatter  <<<NB_EDGE, 256, 0, stream>>>(src, dst, Y, dinv, X);
        gcn_bias_relu<<<NB_NH, 256, 0, stream>>>(X, bs[l]);
    }

    // global mean pool
    gcn_zerof<<<(N_GRAPHS * HIDDEN + N_GRAPHS + 255) / 256, 256, 0, stream>>>(
        pooled, N_GRAPHS * HIDDEN + N_GRAPHS);
    gcn_pool<<<NB_NH, 256, 0, stream>>>(batch, X, pooled);
    gcn_cnt <<<NB_N, 256, 0, stream>>>(batch, cnt);
    gcn_avg <<<(N_GRAPHS * HIDDEN) / 256, 256, 0, stream>>>(pooled, cnt);

    // final linear: Y[0:64,128] = pooled @ Wp ; out = Y + bp
    gcn_gemm_wmma<<<N_GRAPHS / 16, 256, 0, stream>>>(pooled, Wp, Y, HIDDEN);
    gcn_outbias  <<<(N_GRAPHS * HIDDEN) / 256, 256, 0, stream>>>(Y, bp, (float*)d_out);
}